// DeeperGCN_55705725829884
// MI455X (gfx1250) — compile-verified
//
#include <hip/hip_runtime.h>
#include <math.h>

#define MSG_EPS 1e-7f
#define WAVES 4

typedef __attribute__((ext_vector_type(2))) float v2f;
typedef __attribute__((ext_vector_type(8))) float v8f;

// ---------------------------------------------------------------- atomics
__device__ __forceinline__ void atomic_max_f32(float* addr, float v) {
    // works for mixed signs when buffer initialized to -inf
    if (v >= 0.0f) atomicMax((int*)addr, __float_as_int(v));
    else           atomicMin((unsigned int*)addr, __float_as_uint(v));
}

// ---------------------------------------------------------------- init m/den/num
__global__ void init_buf(float* __restrict__ m, float* __restrict__ den,
                         float* __restrict__ num, int count) {
    int i = blockIdx.x * blockDim.x + threadIdx.x;
    if (i < count) {
        m[i]   = -__builtin_inff();
        den[i] = 0.0f;
        num[i] = 0.0f;
    }
}

// ---------------------------------------------------------------- node encoder: h = x @ node_W + node_b
__global__ void node_enc(const float* __restrict__ x, const float* __restrict__ W,
                         const float* __restrict__ b, float* __restrict__ h, int n_nodes) {
    int i = blockIdx.x * blockDim.x + threadIdx.x;
    if (i >= n_nodes * 64) return;
    int n = i >> 6, f = i & 63;
    float acc = b[f];
#pragma unroll
    for (int k = 0; k < 8; ++k) acc = fmaf(x[n * 8 + k], W[k * 64 + f], acc);
    h[i] = acc;
}

// ---------------------------------------------------------------- edge pass 1: segment max of s = t*(relu(h[src]+ea)+eps)
__global__ void edge_pass1(const float* __restrict__ h, const float* __restrict__ edge_attr,
                           const int* __restrict__ edge_index, const float* __restrict__ edge_W,
                           const float* __restrict__ edge_b, const float* __restrict__ t, int layer,
                           float* __restrict__ mbuf, int nE) {
    int e    = (blockIdx.x * blockDim.x + threadIdx.x) >> 5;
    int lane = threadIdx.x & 31;
    if (e >= nE) return;
    int src = edge_index[e];
    int dst = edge_index[nE + e];
    float ti = t[layer];
    float a[8];
#pragma unroll
    for (int k = 0; k < 8; ++k) a[k] = edge_attr[e * 8 + k];
#pragma unroll
    for (int half = 0; half < 2; ++half) {
        int f = lane + half * 32;
        float ea = edge_b[f];
#pragma unroll
        for (int k = 0; k < 8; ++k) ea = fmaf(a[k], edge_W[k * 64 + f], ea);
        float msg = fmaxf(h[src * 64 + f] + ea, 0.0f) + MSG_EPS;
        atomic_max_f32(&mbuf[dst * 64 + f], msg * ti);
    }
}

// ---------------------------------------------------------------- edge pass 2: den += exp(s-m); num += exp(s-m)*msg
__global__ void edge_pass2(const float* __restrict__ h, const float* __restrict__ edge_attr,
                           const int* __restrict__ edge_index, const float* __restrict__ edge_W,
                           const float* __restrict__ edge_b, const float* __restrict__ t, int layer,
                           const float* __restrict__ mbuf, float* __restrict__ den,
                           float* __restrict__ num, int nE) {
    int e    = (blockIdx.x * blockDim.x + threadIdx.x) >> 5;
    int lane = threadIdx.x & 31;
    if (e >= nE) return;
    int src = edge_index[e];
    int dst = edge_index[nE + e];
    float ti = t[layer];
    float a[8];
#pragma unroll
    for (int k = 0; k < 8; ++k) a[k] = edge_attr[e * 8 + k];
#pragma unroll
    for (int half = 0; half < 2; ++half) {
        int f = lane + half * 32;
        float ea = edge_b[f];
#pragma unroll
        for (int k = 0; k < 8; ++k) ea = fmaf(a[k], edge_W[k * 64 + f], ea);
        float msg = fmaxf(h[src * 64 + f] + ea, 0.0f) + MSG_EPS;
        float s   = msg * ti;
        float ex  = __expf(s - mbuf[dst * 64 + f]);
        atomicAdd(&den[dst * 64 + f], ex);
        atomicAdd(&num[dst * 64 + f], ex * msg);
    }
}

// ---------------------------------------------------------------- fused per-layer node MLP (WMMA f32 16x16x4)
// wave-private 16-node tile; A-frag layout: lane = M + 16*(K>>1), vgpr = K&1
__global__ __launch_bounds__(128) void node_mlp(
    float* __restrict__ h, const float* __restrict__ numb, const float* __restrict__ denb,
    const float* __restrict__ W1, const float* __restrict__ b1,
    const float* __restrict__ g1, const float* __restrict__ bt1,
    const float* __restrict__ W2, const float* __restrict__ b2,
    const float* __restrict__ ng, const float* __restrict__ nb, int n_nodes) {
    __shared__ float lds_a[WAVES][16 * 65];   // out tile (stage1 A), reused for z2 tile
    __shared__ float lds_z[WAVES][16 * 129];  // z1 tile
    int lane = threadIdx.x & 31;
    int wv   = threadIdx.x >> 5;
    int tile = blockIdx.x * WAVES + wv;
    if (tile * 16 >= n_nodes) return;
    int row = lane & 15;
    int hi  = lane >> 4;
    float* so = lds_a[wv];
    float* sz = lds_z[wv];
    int base  = tile * 16;

    // stage0: out = h + agg (agg = num/den, 0 for empty segments)
    for (int idx = lane; idx < 16 * 64; idx += 32) {
        int r = idx >> 6, c = idx & 63;
        int gi   = (base + r) * 64 + c;
        float d  = denb[gi];
        float ag = (d > 0.0f) ? (numb[gi] / d) : 0.0f;
        so[r * 65 + c] = h[gi] + ag;
    }

    // stage1: z1[16,128] = out[16,64] @ W1[64,128] + b1
    for (int nt = 0; nt < 8; ++nt) {
        int col  = nt * 16 + row;
        float bb = b1[col];
        v8f acc  = {bb, bb, bb, bb, bb, bb, bb, bb};
#pragma unroll
        for (int kk = 0; kk < 16; ++kk) {
            int k0 = kk * 4 + 2 * hi;
            v2f av, bv;
            av.x = so[row * 65 + k0];
            av.y = so[row * 65 + k0 + 1];
            bv.x = W1[k0 * 128 + col];
            bv.y = W1[(k0 + 1) * 128 + col];
            acc = __builtin_amdgcn_wmma_f32_16x16x4_f32(false, av, false, bv,
                                                        (short)0, acc, false, false);
        }
#pragma unroll
        for (int v = 0; v < 8; ++v) sz[(v + 8 * hi) * 129 + col] = acc[v];
    }

    // LayerNorm(128) + ReLU on z1 (row-wise; halves combined with shfl_xor 16)
    {
        float s = 0.0f;
        for (int c = 0; c < 64; ++c) s += sz[row * 129 + hi * 64 + c];
        s += __shfl_xor(s, 16, 32);
        float mu = s * (1.0f / 128.0f);
        float vs = 0.0f;
        for (int c = 0; c < 64; ++c) {
            float d = sz[row * 129 + hi * 64 + c] - mu;
            vs = fmaf(d, d, vs);
        }
        vs += __shfl_xor(vs, 16, 32);
        float rinv = rsqrtf(vs * (1.0f / 128.0f) + 1e-5f);
        for (int c = 0; c < 64; ++c) {
            int col  = hi * 64 + c;
            float vv = (sz[row * 129 + col] - mu) * rinv * g1[col] + bt1[col];
            sz[row * 129 + col] = fmaxf(vv, 0.0f);
        }
    }

    // stage2: z2[16,64] = z1n[16,128] @ W2[128,64] + b2
    for (int nt = 0; nt < 4; ++nt) {
        int col  = nt * 16 + row;
        float bb = b2[col];
        v8f acc  = {bb, bb, bb, bb, bb, bb, bb, bb};
#pragma unroll
        for (int kk = 0; kk < 32; ++kk) {
            int k0 = kk * 4 + 2 * hi;
            v2f av, bv;
            av.x = sz[row * 129 + k0];
            av.y = sz[row * 129 + k0 + 1];
            bv.x = W2[k0 * 64 + col];
            bv.y = W2[(k0 + 1) * 64 + col];
            acc = __builtin_amdgcn_wmma_f32_16x16x4_f32(false, av, false, bv,
                                                        (short)0, acc, false, false);
        }
#pragma unroll
        for (int v = 0; v < 8; ++v) so[(v + 8 * hi) * 65 + col] = acc[v];
    }

    // LayerNorm(64) + ReLU + residual into h
    {
        float s = 0.0f;
        for (int c = 0; c < 32; ++c) s += so[row * 65 + hi * 32 + c];
        s += __shfl_xor(s, 16, 32);
        float mu = s * (1.0f / 64.0f);
        float vs = 0.0f;
        for (int c = 0; c < 32; ++c) {
            float d = so[row * 65 + hi * 32 + c] - mu;
            vs = fmaf(d, d, vs);
        }
        vs += __shfl_xor(vs, 16, 32);
        float rinv = rsqrtf(vs * (1.0f / 64.0f) + 1e-5f);
        for (int c = 0; c < 32; ++c) {
            int col  = hi * 32 + c;
            float vv = (so[row * 65 + col] - mu) * rinv * ng[col] + nb[col];
            vv       = fmaxf(vv, 0.0f);
            int gi   = (base + row) * 64 + col;
            h[gi]    = h[gi] + vv;
        }
    }
}

// ---------------------------------------------------------------- final: out[N,112] = h @ lin_W + lin_b (WMMA)
__global__ __launch_bounds__(128) void final_lin(const float* __restrict__ h,
                                                 const float* __restrict__ Wl,
                                                 const float* __restrict__ bl,
                                                 float* __restrict__ out, int n_nodes) {
    __shared__ float lds_h[WAVES][16 * 65];
    int lane = threadIdx.x & 31;
    int wv   = threadIdx.x >> 5;
    int tile = blockIdx.x * WAVES + wv;
    if (tile * 16 >= n_nodes) return;
    int row = lane & 15;
    int hi  = lane >> 4;
    float* sh = lds_h[wv];
    int base  = tile * 16;
    for (int idx = lane; idx < 16 * 64; idx += 32) {
        int r = idx >> 6, c = idx & 63;
        sh[r * 65 + c] = h[(base + r) * 64 + c];
    }
    for (int nt = 0; nt < 7; ++nt) {
        int col  = nt * 16 + row;
        float bb = bl[col];
        v8f acc  = {bb, bb, bb, bb, bb, bb, bb, bb};
#pragma unroll
        for (int kk = 0; kk < 16; ++kk) {
            int k0 = kk * 4 + 2 * hi;
            v2f av, bv;
            av.x = sh[row * 65 + k0];
            av.y = sh[row * 65 + k0 + 1];
            bv.x = Wl[k0 * 112 + col];
            bv.y = Wl[(k0 + 1) * 112 + col];
            acc = __builtin_amdgcn_wmma_f32_16x16x4_f32(false, av, false, bv,
                                                        (short)0, acc, false, false);
        }
#pragma unroll
        for (int v = 0; v < 8; ++v) out[(base + v + 8 * hi) * 112 + col] = acc[v];
    }
}

// ---------------------------------------------------------------- launch
extern "C" void kernel_launch(void* const* d_in, const int* in_sizes, int n_in,
                              void* d_out, int out_size, void* d_ws, size_t ws_size,
                              hipStream_t stream) {
    const float* x         = (const float*)d_in[0];
    const float* edge_attr = (const float*)d_in[1];
    const int*   edge_idx  = (const int*)d_in[2];
    const float* node_W    = (const float*)d_in[3];
    const float* node_b    = (const float*)d_in[4];
    const float* edge_W    = (const float*)d_in[5];
    const float* edge_b    = (const float*)d_in[6];
    const float* t         = (const float*)d_in[7];
    const float* W1        = (const float*)d_in[8];
    const float* b1        = (const float*)d_in[9];
    const float* g1        = (const float*)d_in[10];
    const float* bt1       = (const float*)d_in[11];
    const float* W2        = (const float*)d_in[12];
    const float* b2        = (const float*)d_in[13];
    const float* ng        = (const float*)d_in[14];
    const float* nb        = (const float*)d_in[15];
    const float* linW      = (const float*)d_in[16];
    const float* linb      = (const float*)d_in[17];

    int Nn = in_sizes[0] / 8;
    int Ee = in_sizes[1] / 8;
    int L  = in_sizes[7];

    size_t NH  = (size_t)Nn * 64;
    float* h   = (float*)d_ws;
    float* mb  = h + NH;
    float* den = mb + NH;
    float* num = den + NH;

    // node encoder
    {
        int total = Nn * 64;
        node_enc<<<(total + 255) / 256, 256, 0, stream>>>(x, node_W, node_b, h, Nn);
    }

    int edge_blocks = (Ee + 7) / 8;                 // 8 edges (waves) per 256-thread block
    int node_tiles  = (Nn + 15) / 16;
    int mlp_blocks  = (node_tiles + WAVES - 1) / WAVES;
    int init_blocks = ((int)NH + 255) / 256;

    for (int i = 0; i < L; ++i) {
        init_buf<<<init_blocks, 256, 0, stream>>>(mb, den, num, (int)NH);
        edge_pass1<<<edge_blocks, 256, 0, stream>>>(h, edge_attr, edge_idx, edge_W, edge_b,
                                                    t, i, mb, Ee);
        edge_pass2<<<edge_blocks, 256, 0, stream>>>(h, edge_attr, edge_idx, edge_W, edge_b,
                                                    t, i, mb, den, num, Ee);
        node_mlp<<<mlp_blocks, 128, 0, stream>>>(h, num, den,
                                                 W1 + (size_t)i * 64 * 128, b1 + (size_t)i * 128,
                                                 g1 + (size_t)i * 128, bt1 + (size_t)i * 128,
                                                 W2 + (size_t)i * 128 * 64, b2 + (size_t)i * 64,
                                                 ng + (size_t)i * 64, nb + (size_t)i * 64, Nn);
    }
    final_lin<<<mlp_blocks, 128, 0, stream>>>(h, linW, linb, (float*)d_out, Nn);
}